// RGCN_58506044506841
// MI455X (gfx1250) — compile-verified
//
#include <hip/hip_runtime.h>
#include <hip/hip_bf16.h>

typedef __attribute__((ext_vector_type(2))) float v2f;
typedef __attribute__((ext_vector_type(8))) float v8f;

#define RELS  8
#define SLOTS 9   // 8 relations + 1 root slot folded into the GEMM

// ---------------------------------------------------------------------------
// Build W and store it directly in WMMA B-fragment order:
//   Wpack[((ct*(K/4) + kq)*32 + lane)*2 + j]
// where for output column c and row k:
//   ct = c/16, lrow = c%16, kq = k/4, krem = k%4,
//   lane = lrow + 16*(krem/2), j = krem%2.
// This makes the GEMM's per-k-step B load a single coalesced b64.
// Logical value: W[k][r*Cout+o] = sum_b comp[r][b]*basis[b][k][o] (r<8),
//                W[k][8*Cout+o] = root[k][o].
// ---------------------------------------------------------------------------
__global__ void build_W_kernel(const float* __restrict__ basis,
                               const float* __restrict__ comp,
                               const float* __restrict__ root,
                               float* __restrict__ Wpack, int K, int Cout) {
    int idx = blockIdx.x * blockDim.x + threadIdx.x;
    int C = SLOTS * Cout;
    if (idx >= K * C) return;
    int k = idx / C, c = idx - k * C;
    int r = c / Cout, o = c - r * Cout;
    float v;
    if (r < RELS) {
        v = 0.f;
#pragma unroll
        for (int b = 0; b < RELS; ++b)
            v = fmaf(comp[r * RELS + b], basis[((size_t)b * K + k) * Cout + o], v);
    } else {
        v = root[(size_t)k * Cout + o];
    }
    // scatter into B-fragment layout
    int ct = c >> 4, lrow = c & 15;
    int kq = k >> 2, krem = k & 3;
    int lane = lrow + ((krem >> 1) << 4);
    int j = krem & 1;
    Wpack[(((size_t)ct * (K / 4) + kq) * 32 + lane) * 2 + j] = v;
}

// ---------------------------------------------------------------------------
// Y[N, C] = X[N, K] @ W[K, C] using V_WMMA_F32_16X16X4_F32 (full fp32).
// Block = 256 threads = 8 wave32s; block stages a 16-row tile of X in LDS,
// each wave walks 16-wide column tiles.
// A layout (ISA 7.12.2, 32-bit A 16x4): lanes 0-15 rows M, VGPR0/1 = K0/K1;
// lanes 16-31 same rows, VGPR0/1 = K2/K3  -> one aligned ds_load_b64/lane.
// B comes pre-packed per fragment   -> one coalesced global_load_b64/lane
// with immediate +256B offsets inside an s_clause.
// C/D: VGPR v -> row v (lanes<16) or row v+8 (lanes>=16), col = lane%16.
// ---------------------------------------------------------------------------
template <int K>
__global__ void gemm_wmma_kernel(const float* __restrict__ X,
                                 const float* __restrict__ Wpack,
                                 float* __restrict__ Y,
                                 int nrows, int C) {
    constexpr int LDA = K + 4;               // pad: 16 rows hit 16 distinct banks
    constexpr int KQ  = K / 4;
    __shared__ float lds[16 * LDA];
    const int tid  = threadIdx.x;
    const int row0 = blockIdx.x * 16;

    // cooperative float4 load of the 16 x K A tile
    for (int i = tid * 4; i < 16 * K; i += blockDim.x * 4) {
        int r = i / K, k = i - r * K;
        float4 v = {0.f, 0.f, 0.f, 0.f};
        if (row0 + r < nrows)
            v = *(const float4*)(X + (size_t)(row0 + r) * K + k);
        *(float4*)(&lds[r * LDA + k]) = v;
    }
    __syncthreads();

    const int wave   = tid >> 5;
    const int lane   = tid & 31;
    const int lrow   = lane & 15;            // A row / B,D col within tile
    const int khalf  = (lane >> 4) << 1;     // 0 or 2
    const int rowoff = (lane >> 4) << 3;     // 0 or 8 (D rows)

    const float* abase = &lds[lrow * LDA + khalf];   // 8-byte aligned
    const int ntiles    = C >> 4;
    const bool fulltile = (row0 + 16 <= nrows);      // wave-uniform

    for (int ct = wave; ct < ntiles; ct += 8) {
        const v2f* bp = (const v2f*)Wpack + (size_t)ct * KQ * 32 + lane;
        v8f acc = {};
#pragma unroll 8
        for (int kq = 0; kq < KQ; ++kq) {
            v2f a = *(const v2f*)(abase + kq * 4);   // ds_load_b64
            v2f b = bp[(size_t)kq * 32];             // global_load_b64, +256B/step
            acc = __builtin_amdgcn_wmma_f32_16x16x4_f32(
                false, a, false, b, (short)0, acc, false, false);
        }
        const int col = (ct << 4) + lrow;
        float* yp = Y + (size_t)(row0 + rowoff) * C + col;
        if (fulltile) {
#pragma unroll
            for (int v = 0; v < 8; ++v) yp[(size_t)v * C] = acc[v];
        } else {
#pragma unroll
            for (int v = 0; v < 8; ++v)
                if (row0 + rowoff + v < nrows) yp[(size_t)v * C] = acc[v];
        }
    }
}

// ---------------------------------------------------------------------------
// Graph-only degree normalization: cnt[dst*8 + etype] += 1, then invert.
// ---------------------------------------------------------------------------
__global__ void edge_count_kernel(const int* __restrict__ dst,
                                  const int* __restrict__ et,
                                  float* __restrict__ cnt, int E) {
    int e = blockIdx.x * blockDim.x + threadIdx.x;
    if (e < E) atomicAdd(&cnt[(size_t)dst[e] * RELS + et[e]], 1.0f);
}

__global__ void invert_kernel(float* __restrict__ cnt, int n) {
    int i = blockIdx.x * blockDim.x + threadIdx.x;
    if (i < n) cnt[i] = 1.0f / fmaxf(cnt[i], 1.0f);
}

// ---------------------------------------------------------------------------
// agg[dst, o..o+1] += xw[src, etype, o..o+1] * inv[dst, etype]
// One thread per (edge, channel pair): b64 gather + 2 f32 atomics.
// ---------------------------------------------------------------------------
__global__ void edge_scatter_kernel(const int* __restrict__ src,
                                    const int* __restrict__ dst,
                                    const int* __restrict__ et,
                                    const float* __restrict__ xw,
                                    const float* __restrict__ inv,
                                    float* __restrict__ agg,
                                    int E, int Cout) {
    const int half = Cout >> 1;
    size_t t = (size_t)blockIdx.x * blockDim.x + threadIdx.x;
    size_t total = (size_t)E * half;
    if (t >= total) return;
    int e = (int)(t / half);
    int o = (int)(t - (size_t)e * half) << 1;
    int s = src[e], d = dst[e], r = et[e];
    float sc = inv[(size_t)d * RELS + r];
    float2 m = *(const float2*)&xw[((size_t)s * SLOTS + r) * Cout + o];
    float* ap = &agg[(size_t)d * Cout + o];
    atomicAdd(ap,     m.x * sc);
    atomicAdd(ap + 1, m.y * sc);
}

// ---------------------------------------------------------------------------
// h = [relu]( agg + xw[:, root_slot, :] + bias )
// ---------------------------------------------------------------------------
__global__ void epilogue_kernel(const float* __restrict__ agg,
                                const float* __restrict__ xw,
                                const float* __restrict__ bias,
                                float* __restrict__ h,
                                int n, int Cout, int do_relu) {
    int idx = blockIdx.x * blockDim.x + threadIdx.x;
    if (idx >= n * Cout) return;
    int node = idx / Cout, o = idx - node * Cout;
    float v = agg[idx] + xw[((size_t)node * SLOTS + RELS) * Cout + o] + bias[o];
    h[idx] = do_relu ? fmaxf(v, 0.f) : v;
}

// ---------------------------------------------------------------------------
extern "C" void kernel_launch(void* const* d_in, const int* in_sizes, int n_in,
                              void* d_out, int out_size, void* d_ws, size_t ws_size,
                              hipStream_t stream) {
    const float* x      = (const float*)d_in[0];
    const int*   eidx   = (const int*)d_in[1];
    const int*   etype  = (const int*)d_in[2];
    const float* basis1 = (const float*)d_in[3];
    const float* comp1  = (const float*)d_in[4];
    const float* root1  = (const float*)d_in[5];
    const float* bias1  = (const float*)d_in[6];
    const float* basis2 = (const float*)d_in[7];
    const float* comp2  = (const float*)d_in[8];
    const float* root2  = (const float*)d_in[9];
    const float* bias2  = (const float*)d_in[10];
    const float* basis3 = (const float*)d_in[11];
    const float* comp3  = (const float*)d_in[12];
    const float* root3  = (const float*)d_in[13];
    const float* bias3  = (const float*)d_in[14];

    const int IN_C = 128, HID = 64, OUT = 32;
    const int N = in_sizes[0] / IN_C;   // 100000
    const int E = in_sizes[2];          // 1600000
    const int* src = eidx;
    const int* dst = eidx + E;

    // workspace carve-up (floats)
    float* ws   = (float*)d_ws;
    float* xw   = ws;                                   // N * 9 * 64 (max)
    float* agg  = xw   + (size_t)N * SLOTS * HID;       // N * 64
    float* h    = agg  + (size_t)N * HID;               // N * 64 (reused)
    float* Wall = h    + (size_t)N * HID;               // 128 * 9 * 64
    float* invn = Wall + (size_t)IN_C * SLOTS * HID;    // N * 8

    // --- per-(node, relation) mean normalization: graph-only, computed once
    hipMemsetAsync(invn, 0, (size_t)N * RELS * sizeof(float), stream);
    edge_count_kernel<<<(E + 255) / 256, 256, 0, stream>>>(dst, etype, invn, E);
    {
        int tot = N * RELS;
        invert_kernel<<<(tot + 255) / 256, 256, 0, stream>>>(invn, tot);
    }

    // ---------------- Layer 1: 128 -> 64, relu ----------------
    {
        const int K = 128, Cout = HID, C = SLOTS * Cout;
        build_W_kernel<<<(K * C + 255) / 256, 256, 0, stream>>>(basis1, comp1, root1, Wall, K, Cout);
        gemm_wmma_kernel<128><<<(N + 15) / 16, 256, 0, stream>>>(x, Wall, xw, N, C);
        hipMemsetAsync(agg, 0, (size_t)N * Cout * sizeof(float), stream);
        size_t tot = (size_t)E * (Cout / 2);
        edge_scatter_kernel<<<(unsigned)((tot + 255) / 256), 256, 0, stream>>>(
            src, dst, etype, xw, invn, agg, E, Cout);
        int nt = N * Cout;
        epilogue_kernel<<<(nt + 255) / 256, 256, 0, stream>>>(agg, xw, bias1, h, N, Cout, 1);
    }

    // ---------------- Layer 2: 64 -> 64, relu ----------------
    {
        const int K = 64, Cout = HID, C = SLOTS * Cout;
        build_W_kernel<<<(K * C + 255) / 256, 256, 0, stream>>>(basis2, comp2, root2, Wall, K, Cout);
        gemm_wmma_kernel<64><<<(N + 15) / 16, 256, 0, stream>>>(h, Wall, xw, N, C);
        hipMemsetAsync(agg, 0, (size_t)N * Cout * sizeof(float), stream);
        size_t tot = (size_t)E * (Cout / 2);
        edge_scatter_kernel<<<(unsigned)((tot + 255) / 256), 256, 0, stream>>>(
            src, dst, etype, xw, invn, agg, E, Cout);
        int nt = N * Cout;
        epilogue_kernel<<<(nt + 255) / 256, 256, 0, stream>>>(agg, xw, bias2, h, N, Cout, 1);
    }

    // ---------------- Layer 3: 64 -> 32, no relu, to d_out ----------------
    {
        const int K = 64, Cout = OUT, C = SLOTS * Cout;
        build_W_kernel<<<(K * C + 255) / 256, 256, 0, stream>>>(basis3, comp3, root3, Wall, K, Cout);
        gemm_wmma_kernel<64><<<(N + 15) / 16, 256, 0, stream>>>(h, Wall, xw, N, C);
        hipMemsetAsync(agg, 0, (size_t)N * Cout * sizeof(float), stream);
        size_t tot = (size_t)E * (Cout / 2);
        edge_scatter_kernel<<<(unsigned)((tot + 255) / 256), 256, 0, stream>>>(
            src, dst, etype, xw, invn, agg, E, Cout);
        int nt = N * Cout;
        epilogue_kernel<<<(nt + 255) / 256, 256, 0, stream>>>(
            agg, xw, bias3, (float*)d_out, N, Cout, 0);
    }
}